// KbModel_7017976561923
// MI455X (gfx1250) — compile-verified
//
#include <hip/hip_runtime.h>
#include <math.h>

typedef float v2f __attribute__((ext_vector_type(2)));

#define NSTEPS 100

// ---- compile-time Adam bias-correction constants, pre-splatted pairs ----
// (single uniform s_load_b64 per constant per step; both halves equal)
struct AdamC {
  alignas(8) float c1[NSTEPS][2];
  alignas(8) float c2[NSTEPS][2];
};
static constexpr AdamC make_adam() {
  AdamC a{};
  double b1t = 1.0, b2t = 1.0;
  for (int t = 0; t < NSTEPS; ++t) {
    b1t *= 0.9; b2t *= 0.999;
    const float c1 = (float)(0.01 / (1.0 - b1t));  // LR / (1 - B1^t)
    const float c2 = (float)(1.0  / (1.0 - b2t));  // 1  / (1 - B2^t)
    a.c1[t][0] = c1; a.c1[t][1] = c1;
    a.c2[t][0] = c2; a.c2[t][1] = c2;
  }
  return a;
}
__constant__ AdamC g_adam = make_adam();

// ---- guaranteed packed-FP32 VOP3P ops (CDNA5 v_pk_*_f32, 2 f32/lane) ----
// NOTE: gfx1250 assembler rejects SGPR-pair sources on packed-f32 VOP3P
// ("invalid op_sel operand"), so all sources are VGPR pairs.
static __device__ __forceinline__ v2f vsplat(float s) { return (v2f){s, s}; }
static __device__ __forceinline__ v2f pk_fma(v2f a, v2f b, v2f c) {
  v2f d;
  asm("v_pk_fma_f32 %0, %1, %2, %3" : "=v"(d) : "v"(a), "v"(b), "v"(c));
  return d;
}
static __device__ __forceinline__ v2f pk_mul(v2f a, v2f b) {
  v2f d;
  asm("v_pk_mul_f32 %0, %1, %2" : "=v"(d) : "v"(a), "v"(b));
  return d;
}
static __device__ __forceinline__ v2f pk_add(v2f a, v2f b) {
  v2f d;
  asm("v_pk_add_f32 %0, %1, %2" : "=v"(d) : "v"(a), "v"(b));
  return d;
}
// trans-pipe ops (co-execute with the VALU pk stream)
static __device__ __forceinline__ v2f vsqrt2(v2f a) {
  return (v2f){__builtin_amdgcn_sqrtf(a.x), __builtin_amdgcn_sqrtf(a.y)};
}
static __device__ __forceinline__ v2f vrcp2(v2f a) {
  return (v2f){__builtin_amdgcn_rcpf(a.x), __builtin_amdgcn_rcpf(a.y)};
}

// ---- 100-step Adam solve for theta, P independent packed chains ----
template <int P>
static __device__ __forceinline__ void adam_theta(
    v2f (&TH)[P], const v2f (&RU)[P],
    float k0, float k1, float k2, float k3, float k4,
    float d1, float d2, float d3, float d4, float twon)
{
  const v2f vk4 = vsplat(k4), vk3 = vsplat(k3), vk2 = vsplat(k2);
  const v2f vk1 = vsplat(k1), vk0 = vsplat(k0);
  const v2f vd4 = vsplat(d4), vd3 = vsplat(d3), vd2 = vsplat(d2), vd1 = vsplat(d1);
  const v2f vtw = vsplat(twon), vm2 = vsplat(-2.0f), veps = vsplat(1e-8f);
  const v2f vb1 = vsplat(0.9f),   vob1 = vsplat(0.1f);
  const v2f vb2 = vsplat(0.999f), vob2 = vsplat(0.001f);

  v2f M[P], V[P], NTR[P];
#pragma unroll
  for (int p = 0; p < P; ++p) {
    TH[p] = vsplat(0.f); M[p] = vsplat(0.f); V[p] = vsplat(0.f);
    NTR[p] = pk_mul(vsplat(-twon), RU[p]);   // -(2^-21 * r): exact (pow-2 scale)
  }

#pragma unroll 4
  for (int t = 0; t < NSTEPS; ++t) {
    const v2f vc1 = *(const v2f*)g_adam.c1[t];   // uniform s_load_b64 -> 2 movs
    const v2f vc2 = *(const v2f*)g_adam.c2[t];
#pragma unroll
    for (int p = 0; p < P; ++p) {
      v2f th = TH[p];
      // f(theta), f'(theta): Horner chains of v_pk_fma_f32
      v2f f  = pk_fma(vk4, th, vk3);
      f      = pk_fma(f,  th, vk2);
      f      = pk_fma(f,  th, vk1);
      f      = pk_fma(f,  th, vk0);
      v2f fp = pk_fma(vd4, th, vd3);
      fp     = pk_fma(fp, th, vd2);
      fp     = pk_fma(fp, th, vd1);
      v2f s  = pk_fma(vtw, f, NTR[p]);       // == twon*(f - r) bit-exact
      v2f g  = pk_mul(s, fp);
      v2f m  = pk_fma(vb1, M[p], pk_mul(vob1, g));
      v2f vv = pk_fma(vb2, V[p], pk_mul(vob2, pk_mul(g, g)));
      M[p] = m; V[p] = vv;
      v2f den = pk_add(vsqrt2(pk_mul(vc2, vv)), veps);  // sqrt(v_hat)+eps
      v2f rc  = vrcp2(den);
      v2f tt  = pk_fma(den, rc, vm2);        // den*rc - 2
      v2f rcn = pk_mul(rc, tt);              // = -(Newton-refined 1/den)
      v2f num = pk_mul(vc1, m);              // lr/(1-b1^t) * m
      TH[p]   = pk_fma(num, rcn, th);        // th - num/den
    }
  }
}

// ---- once-per-element re-projection epilogue (precise libm) ----
static __device__ __forceinline__ void project_uv(
    float mx, float my, float ru, float theta,
    float k0, float k1, float k2, float k3, float k4,
    float& u, float& v)
{
  float s, c;
  sincosf(theta, &s, &c);                    // shared argument reduction
  float px = s * (mx / ru);
  float py = s * (my / ru);
  float rr = sqrtf(px * px + py * py);
  float th = atan2f(rr, c);
  float d  = fmaf(fmaf(fmaf(fmaf(k4, th, k3), th, k2), th, k1), th, k0);
  u = fmaf(d * px, 600.0f, 512.0f);
  v = fmaf(d * py, 600.0f, 512.0f);
}

__global__ __launch_bounds__(256) void fisheye_invert_kernel(
    const float* __restrict__ in, const float* __restrict__ kv,
    float* __restrict__ out, int npts, float twon)
{
  __shared__ float4 tile[512];                 // 8 KB: 1024 points per block
  const int tid  = threadIdx.x;
  const int blk  = blockIdx.x;
  const int base = blk * 1024;                 // first point of this block

  const float k0 = kv[0], k1 = kv[1], k2 = kv[2], k3 = kv[3], k4 = kv[4];
  const float d1 = k1, d2 = 2.0f * k2, d3 = 3.0f * k3, d4 = 4.0f * k4;

  if (base + 1024 <= npts) {
    // ---- CDNA5 async copy: HBM -> LDS (ASYNCcnt), 32 B/lane contiguous ----
    const float4* gp0 = (const float4*)in + (size_t)blk * 512 + 2 * tid;
    __builtin_prefetch(gp0, 0, 3);             // global_prefetch_b8
    unsigned l0 = (unsigned)(size_t)(void*)&tile[2 * tid];      // LDS byte addr
    unsigned l1 = (unsigned)(size_t)(void*)&tile[2 * tid + 1];
    asm volatile("global_load_async_to_lds_b128 %0, %1, off" :: "v"(l0), "v"(gp0)     : "memory");
    asm volatile("global_load_async_to_lds_b128 %0, %1, off" :: "v"(l1), "v"(gp0 + 1) : "memory");
    asm volatile("s_wait_asynccnt 0" ::: "memory");
    __syncthreads();

    float4 a = tile[2 * tid];                  // (x0,y0,x1,y1)
    float4 b = tile[2 * tid + 1];              // (x2,y2,x3,y3)

    v2f mx[2], my[2], ru[2], th[2];
    mx[0] = (v2f){(a.x - 512.0f) / 600.0f, (a.z - 512.0f) / 600.0f};
    my[0] = (v2f){(a.y - 512.0f) / 600.0f, (a.w - 512.0f) / 600.0f};
    mx[1] = (v2f){(b.x - 512.0f) / 600.0f, (b.z - 512.0f) / 600.0f};
    my[1] = (v2f){(b.y - 512.0f) / 600.0f, (b.w - 512.0f) / 600.0f};
#pragma unroll
    for (int p = 0; p < 2; ++p) {
      ru[p].x = sqrtf(mx[p].x * mx[p].x + my[p].x * my[p].x);
      ru[p].y = sqrtf(mx[p].y * mx[p].y + my[p].y * my[p].y);
    }

    adam_theta<2>(th, ru, k0, k1, k2, k3, k4, d1, d2, d3, d4, twon);

    float u0, v0, u1, v1, u2, v2, u3, v3;
    project_uv(mx[0].x, my[0].x, ru[0].x, th[0].x, k0, k1, k2, k3, k4, u0, v0);
    project_uv(mx[0].y, my[0].y, ru[0].y, th[0].y, k0, k1, k2, k3, k4, u1, v1);
    project_uv(mx[1].x, my[1].x, ru[1].x, th[1].x, k0, k1, k2, k3, k4, u2, v2);
    project_uv(mx[1].y, my[1].y, ru[1].y, th[1].y, k0, k1, k2, k3, k4, u3, v3);

    float4* gout = (float4*)out + (size_t)blk * 512 + 2 * tid;
    gout[0] = make_float4(u0, v0, u1, v1);
    gout[1] = make_float4(u2, v2, u3, v3);
  } else {
    // generic tail (not taken for N = 4M): one element at a time
    for (int e = base + tid; e < npts; e += 256) {
      float x = in[2 * e], y = in[2 * e + 1];
      float mxs = (x - 512.0f) / 600.0f;
      float mys = (y - 512.0f) / 600.0f;
      float rus = sqrtf(mxs * mxs + mys * mys);
      v2f RU[1] = {vsplat(rus)}, TH[1];
      adam_theta<1>(TH, RU, k0, k1, k2, k3, k4, d1, d2, d3, d4, twon);
      float u, v;
      project_uv(mxs, mys, rus, TH[0].x, k0, k1, k2, k3, k4, u, v);
      out[2 * e]     = u;
      out[2 * e + 1] = v;
    }
  }
}

extern "C" void kernel_launch(void* const* d_in, const int* in_sizes, int n_in,
                              void* d_out, int out_size, void* d_ws, size_t ws_size,
                              hipStream_t stream) {
  const float* in = (const float*)d_in[0];   // [N,2] float32 pixels
  const float* kv = (const float*)d_in[1];   // [5] float32 k-vector
  float* out = (float*)d_out;                // [N,2] float32 (u,v)
  const int npts  = in_sizes[0] / 2;
  const float twon = 2.0f / (float)npts;     // 2/N (exact: 2^-21)
  const int blocks = (npts + 1023) / 1024;   // 4 points/thread, 256 threads
  fisheye_invert_kernel<<<blocks, 256, 0, stream>>>(in, kv, out, npts, twon);
}